// bias_time_87960930222572
// MI455X (gfx1250) — compile-verified
//
#include <hip/hip_runtime.h>

// ---------------------------------------------------------------------------
// LSTM as one long serial scan (T = B*S = 2,457,600 steps, U = 3, D = 3).
//   Kernel A: xz[T,12] = X[T,3] @ W[3,12] + bias   via v_wmma_f32_16x16x4_f32
//             (one WMMA per 16 timesteps; LDS-staged coalesced b128 writeback)
//   Kernel A': ragged-tail rows (none at T=2457600) via plain VALU
//   Kernel B: single-wave serial recurrence, 1 hw tanh per lane per step,
//             ds_bpermute shuffles for i/f/g/o regather + h broadcast.
// ---------------------------------------------------------------------------

typedef __attribute__((ext_vector_type(2))) float v2f;
typedef __attribute__((ext_vector_type(8))) float v8f;

#if __has_builtin(__builtin_amdgcn_tanhf)
__device__ __forceinline__ float fast_tanh(float x) { return __builtin_amdgcn_tanhf(x); }
#else
__device__ __forceinline__ float fast_tanh(float x) { return tanhf(x); }
#endif

// -------------------------------------------------------------------
// Kernel A: one 16-timestep tile per WMMA. No tail guards in the loop.
//   A (16x4 f32): lanes 0-15 {K0,K1}, lanes 16-31 {K2,K3=0}
//   B (4x16 f32): VGPR0 = K0-row / K2-row, VGPR1 = K1-row / K3(=0)
//   C = bias broadcast; D (16x16): VGPR r -> M = r + 8*hi, N = lane&15
//   Epilogue: stage 16x12 tile (192 contiguous floats) in LDS, then
//   coalesced global b128 + b64 stores.
// -------------------------------------------------------------------
__global__ void lstm_proj_wmma_kernel(const float* __restrict__ x,
                                      const float* __restrict__ W,
                                      const float* __restrict__ bias,
                                      float* __restrict__ xz,
                                      int numTiles) {
  __shared__ __align__(16) float stage[8][192];   // 768 B per wave, 8 waves/block
  const int lane   = threadIdx.x & 31;
  const int waveIn = threadIdx.x >> 5;
  const int wave   = (blockIdx.x * blockDim.x + threadIdx.x) >> 5;
  const int nWaves = (gridDim.x * blockDim.x) >> 5;
  const int n      = lane & 15;       // N column (valid when < 12)
  const bool hi    = lane >= 16;
  float* sm = stage[waveIn];

  // B matrix + bias-as-C: loaded once, reused for every tile.
  float bx = (n < 12) ? W[(hi ? 2 : 0) * 12 + n] : 0.0f;   // K0 / K2 row
  float w1 = (n < 12) ? W[12 + n] : 0.0f;                  // K1 row
  v2f bmat = { bx, hi ? 0.0f : w1 };                       // K3 pad = 0
  const float bv = (n < 12) ? bias[n] : 0.0f;
  v8f cmat = {bv, bv, bv, bv, bv, bv, bv, bv};

  for (int tile = wave; tile < numTiles; tile += nWaves) {
    const size_t base = (size_t)tile * 16;
    // Unconditional b96 read of the lane's row, then selects (no branches).
    const float* xr = x + (base + (size_t)n) * 3;
    const float x0 = xr[0], x1 = xr[1], x2 = xr[2];
    v2f amat = { hi ? x2 : x0, hi ? 0.0f : x1 };

    v8f d = __builtin_amdgcn_wmma_f32_16x16x4_f32(
        /*neg_a=*/false, amat, /*neg_b=*/false, bmat,
        /*c_mod=*/(short)0, cmat, /*reuse_a=*/false, /*reuse_b=*/false);

    // Stage D into LDS row-major: s[m*12 + n], m = r + 8*hi. One exec
    // toggle for all 8 stores; intra-wave DS ops are in-order (no barrier).
    if (n < 12) {
      float* s = sm + (hi ? 8 : 0) * 12 + n;
#pragma unroll
      for (int r = 0; r < 8; ++r) s[r * 12] = d[r];
    }

    // Coalesced writeback of the 192-float tile:
    //   pass 1: 32 lanes x float4 = 128 floats, pass 2: 32 lanes x float2.
    float* g = xz + base * 12;                      // 768-byte aligned
    const float4 v4 = *(const float4*)(sm + lane * 4);
    const float2 v2 = *(const float2*)(sm + 128 + lane * 2);
    ((float4*)g)[lane]         = v4;
    ((float2*)(g + 128))[lane] = v2;
  }
}

// -------------------------------------------------------------------
// Kernel A': ragged tail rows (t in [tstart, T)) -- plain VALU.
// -------------------------------------------------------------------
__global__ void lstm_proj_tail_kernel(const float* __restrict__ x,
                                      const float* __restrict__ W,
                                      const float* __restrict__ bias,
                                      float* __restrict__ xz,
                                      int tstart, int T) {
  const int idx = blockIdx.x * blockDim.x + threadIdx.x;
  const int nel = (T - tstart) * 12;
  if (idx >= nel) return;
  const int t = tstart + idx / 12;
  const int j = idx % 12;
  const float* xr = x + (size_t)t * 3;
  xz[(size_t)t * 12 + j] =
      bias[j] + xr[0] * W[j] + xr[1] * W[12 + j] + xr[2] * W[24 + j];
}

// -------------------------------------------------------------------
// Kernel B: one wave32 runs the whole recurrence.
//   lane j = lane%12 owns gate column j; every lane mirrors state for
//   index lane%3 so the hot loop is divergence-free.
//   sigmoid(z) = 0.5*(1 + tanh(0.5 z))  -> exactly one v_tanh per lane.
// -------------------------------------------------------------------
__global__ void lstm_scan_kernel(const float* __restrict__ xz,  // null => inline proj
                                 const float* __restrict__ x,
                                 const float* __restrict__ W,
                                 const float* __restrict__ bias,
                                 const float* __restrict__ rk,
                                 const float* __restrict__ h0in,
                                 const float* __restrict__ c0in,
                                 float* __restrict__ out,       // [T*3 | hf(3) | cf(3)]
                                 int T) {
  const int lane = threadIdx.x & 31;
  const int j    = lane % 12;          // gate column
  const int lb   = lane % 3;           // state index this lane mirrors
  const bool isG = (j >= 6) && (j < 9);

  const float rk0 = rk[j], rk1 = rk[12 + j], rk2 = rk[24 + j];
  float w0 = 0.0f, w1 = 0.0f, w2 = 0.0f, bj = 0.0f;
  if (!xz) { w0 = W[j]; w1 = W[12 + j]; w2 = W[24 + j]; bj = bias[j]; }

  float h  = h0in[lb];
  float c  = c0in[lb];
  float h0 = h0in[0], h1 = h0in[1], h2 = h0in[2];

  for (int t = 0; t < T; ++t) {
    float zin;
    if (xz) {
      zin = xz[(size_t)t * 12 + j];
      // Consumption is 48 B/step (~1 cacheline per 2.7 steps): hint one
      // line every 2 steps, ~6 KB ahead. Uniform scalar branch.
      if (((t & 1) == 0) && (t + 128 < T))
        __builtin_prefetch(&xz[(size_t)(t + 128) * 12], 0, 0);
    } else {
      const float* xr = x + (size_t)t * 3;
      zin = bj + xr[0] * w0 + xr[1] * w1 + xr[2] * w2;
    }

    const float z = zin + h0 * rk0 + h1 * rk1 + h2 * rk2;
    const float y = fast_tanh(isG ? z : 0.5f * z);
    const float a = isG ? y : __builtin_fmaf(0.5f, y, 0.5f);

    const float iv = __shfl(a, lb,     32);
    const float fv = __shfl(a, lb + 3, 32);
    const float gv = __shfl(a, lb + 6, 32);
    const float ov = __shfl(a, lb + 9, 32);

    c = fv * c + iv * gv;
    h = ov * fast_tanh(c);

    h0 = __shfl(h, 0, 32);
    h1 = __shfl(h, 1, 32);
    h2 = __shfl(h, 2, 32);

    if (lane < 3) out[(size_t)t * 3 + lane] = h;
  }

  if (lane < 3) {
    out[(size_t)T * 3 + lane]     = h;   // hf
    out[(size_t)T * 3 + 3 + lane] = c;   // cf
  }
}

// -------------------------------------------------------------------
extern "C" void kernel_launch(void* const* d_in, const int* in_sizes, int n_in,
                              void* d_out, int out_size, void* d_ws, size_t ws_size,
                              hipStream_t stream) {
  const float* x    = (const float*)d_in[0];  // (B,S,3)
  const float* h0   = (const float*)d_in[1];  // (3,)
  const float* c0   = (const float*)d_in[2];  // (3,)
  const float* W    = (const float*)d_in[3];  // (3,12)
  const float* rk   = (const float*)d_in[4];  // (3,12)
  const float* bias = (const float*)d_in[5];  // (12,)
  float* out = (float*)d_out;

  const int T = in_sizes[0] / 3;              // B*S = 2,457,600
  const int numTiles = T / 16;                // full WMMA tiles (T%16==0 here)
  const int tailStart = numTiles * 16;

  const size_t xzBytes = (size_t)T * 12 * sizeof(float);
  float* xz = (ws_size >= xzBytes) ? (float*)d_ws : nullptr;

  if (xz) {
    // 512 blocks x 256 thr = 4096 waves; ~37 tiles/wave. HBM-bound
    // (147 MB total traffic ~ 6.3 us at 23.3 TB/s).
    lstm_proj_wmma_kernel<<<dim3(512), dim3(256), 0, stream>>>(
        x, W, bias, xz, numTiles);
    if (tailStart < T) {
      const int nel = (T - tailStart) * 12;
      lstm_proj_tail_kernel<<<dim3((nel + 255) / 256), dim3(256), 0, stream>>>(
          x, W, bias, xz, tailStart, T);
    }
  }
  // Serial recurrence: one wave. Same stream => ordered after kernel A.
  lstm_scan_kernel<<<dim3(1), dim3(32), 0, stream>>>(
      xz, x, W, bias, rk, h0, c0, out, T);
}